// encoder_block_10161892623137
// MI455X (gfx1250) — compile-verified
//
#include <hip/hip_runtime.h>
#include <math.h>

#define BB   4
#define LL   384
#define DINC 128
#define DD   128
#define NEGV (-1.0e7f)
#define EPSV (1e-12f)

typedef __attribute__((ext_vector_type(2))) float v2f;
typedef __attribute__((ext_vector_type(8))) float v8f;

__device__ __forceinline__ float wave_sum(float v) {
#pragma unroll
  for (int off = 16; off > 0; off >>= 1) v += __shfl_xor(v, off, 32);
  return v;
}

// ---------------------------------------------------------------------------
// Kernel 0: Q/V projection with fp32 WMMA (D = A[16x4] x B[4x16] + C, K=128).
// One wave per 16x16 output tile; A-fragment shared between Wq and Wv MACs.
// ---------------------------------------------------------------------------
__global__ __launch_bounds__(32) void proj_wmma_kernel(
    const float* __restrict__ X,  const float* __restrict__ Wq,
    const float* __restrict__ bq, const float* __restrict__ Wv,
    const float* __restrict__ bv, float* __restrict__ qraw,
    float* __restrict__ vraw) {
  const int lane = threadIdx.x & 31;
  const int hi   = lane >> 4;   // 0: lanes 0-15, 1: lanes 16-31
  const int l16  = lane & 15;
  const int col0 = blockIdx.x * 16;
  const int row0 = blockIdx.y * 16;

  v8f cq = {};
  v8f cv = {};
  const float* xrow = X + (size_t)(row0 + l16) * DINC + hi * 2;

  for (int k = 0; k < DINC; k += 4) {
    // A 16x4: lanes 0-15 hold K=k,k+1 ; lanes 16-31 hold K=k+2,k+3
    v2f a;
    a.x = xrow[k];
    a.y = xrow[k + 1];
    // B 4x16: v0 holds K=k(+hi), v1 holds K=k+2(+hi), N striped across lanes
    const int kb0 = k + hi;
    const int kb1 = k + 2 + hi;
    v2f bqf, bvf;
    bqf.x = Wq[kb0 * DD + col0 + l16];
    bqf.y = Wq[kb1 * DD + col0 + l16];
    bvf.x = Wv[kb0 * DD + col0 + l16];
    bvf.y = Wv[kb1 * DD + col0 + l16];
    cq = __builtin_amdgcn_wmma_f32_16x16x4_f32(false, a, false, bqf, (short)0, cq, false, false);
    cv = __builtin_amdgcn_wmma_f32_16x16x4_f32(false, a, false, bvf, (short)0, cv, false, false);
  }

  const float biasq = bq[col0 + l16];
  const float biasv = bv[col0 + l16];
#pragma unroll
  for (int r = 0; r < 8; ++r) {
    const int row = row0 + r + hi * 8;  // C/D layout: VGPR r => M=r (lo lanes), M=r+8 (hi lanes)
    qraw[(size_t)row * DD + col0 + l16] = cq[r] + biasq;
    vraw[(size_t)row * DD + col0 + l16] = cv[r] + biasv;
  }
}

// ---------------------------------------------------------------------------
// Kernel 1: in-place row L2 normalize over D=128 (wave per row, float4 lanes)
// ---------------------------------------------------------------------------
__global__ __launch_bounds__(256) void l2n_rows_kernel(float* __restrict__ buf, int nrows) {
  const int wid = threadIdx.x >> 5, lane = threadIdx.x & 31;
  const int row = blockIdx.x * 8 + wid;
  if (row >= nrows) return;
  float4* p = reinterpret_cast<float4*>(buf + (size_t)row * DD);
  float4 v = p[lane];
  const float s  = wave_sum(v.x * v.x + v.y * v.y + v.z * v.z + v.w * v.w);
  const float rs = rsqrtf(fmaxf(s, EPSV));
  v.x *= rs; v.y *= rs; v.z *= rs; v.w *= rs;
  p[lane] = v;
}

// ---------------------------------------------------------------------------
// Kernel 2: SINGLE streaming pass over rel with online softmax.
// Per (b,i) workgroup (8 waves): scores -> out, online (m,s,acc) per wave,
// deterministic cross-wave merge, att -> out, unnormalized emb acc -> ws,
// per-row sum(||base||^2) -> ws.  rel is read exactly once from HBM.
// ---------------------------------------------------------------------------
__global__ __launch_bounds__(256) void fused_attn_kernel(
    const float* __restrict__ q,   const float* __restrict__ rel,
    const int* __restrict__ vlens, float* __restrict__ sc_out,
    float* __restrict__ att_out,   float* __restrict__ n2part,
    float* __restrict__ embacc) {
  __shared__ float qi_s[DD];
  __shared__ float srow_s[LL];
  __shared__ float stage[8][DD];
  __shared__ float m_s[8];
  __shared__ float s_s[8];
  __shared__ float wn2_s[8];
  __shared__ float red_s[2];

  const int b = blockIdx.y, i = blockIdx.x;
  const int tid = threadIdx.x, lane = tid & 31, wid = tid >> 5;
  const size_t bi = (size_t)(b * LL + i);
  const int vl = vlens[b];

  if (tid < DD) qi_s[tid] = q[bi * DD + tid];
  __syncthreads();

  const float4 qi4 = reinterpret_cast<const float4*>(qi_s)[lane];
  const float* relrow = rel + bi * (size_t)(LL * DD);

  float m = -3.4e38f, s = 0.f;               // online softmax state (wave-uniform)
  float ax = 0.f, ay = 0.f, az = 0.f, aw = 0.f;  // sum of e_j * rel_ij (lane = 4 dims)
  float wn2 = 0.f;

  for (int j = wid; j < LL; j += 8) {
    const float4 r4  = reinterpret_cast<const float4*>(relrow + j * DD)[lane];
    const float4 qj4 = reinterpret_cast<const float4*>(q + (size_t)(b * LL + j) * DD)[lane];
    __builtin_prefetch((const void*)(relrow + (j + 8) * DD + 4 * lane), 0, 1);

    const float bx = qi4.x + r4.x, by = qi4.y + r4.y;
    const float bz = qi4.z + r4.z, bw = qi4.w + r4.w;
    float n2p = bx * bx + by * by + bz * bz + bw * bw;
    float dp  = qj4.x * bx + qj4.y * by + qj4.z * bz + qj4.w * bw;
    n2p = wave_sum(n2p);   // all lanes get totals
    dp  = wave_sum(dp);

    const float sc = dp * rsqrtf(fmaxf(n2p, EPSV));
    if (lane == 0) {
      sc_out[bi * LL + j] = sc;  // raw (pre-mask) scores output
      srow_s[j] = sc;
      wn2 += n2p;                // contribution to global sum-of-squares of `base`
    }

    // online softmax update (masked value; wave-uniform arithmetic)
    const float val   = (j < vl) ? sc : NEGV;
    const float nm    = fmaxf(m, val);
    const float scale = __expf(m - nm);
    const float e     = __expf(val - nm);
    s  = fmaf(s, scale, e);
    ax = fmaf(ax, scale, e * r4.x);
    ay = fmaf(ay, scale, e * r4.y);
    az = fmaf(az, scale, e * r4.z);
    aw = fmaf(aw, scale, e * r4.w);
    m  = nm;
  }

  if (lane == 0) { m_s[wid] = m; s_s[wid] = s; wn2_s[wid] = wn2; }
  __syncthreads();

  if (tid == 0) {  // deterministic fixed-order merges
    float M = m_s[0];
    for (int w = 1; w < 8; ++w) M = fmaxf(M, m_s[w]);
    red_s[0] = M;
    float t = 0.f;
    for (int w = 0; w < 8; ++w) t += wn2_s[w];
    n2part[bi] = t;
  }
  __syncthreads();
  const float M = red_s[0];

  // rescale per-wave partial sums/accumulators to the common max M
  const float wscale = __expf(m - M);  // wave-uniform
  stage[wid][4 * lane + 0] = ax * wscale;
  stage[wid][4 * lane + 1] = ay * wscale;
  stage[wid][4 * lane + 2] = az * wscale;
  stage[wid][4 * lane + 3] = aw * wscale;
  if (lane == 0) s_s[wid] = s * wscale;
  __syncthreads();

  if (tid == 0) {
    float S = 0.f;
    for (int w = 0; w < 8; ++w) S += s_s[w];
    red_s[1] = S;
  }
  __syncthreads();
  const float S   = red_s[1];
  const float inv = 1.0f / S;

  // attention-weight output recomputed from the cached scores row
  for (int j = tid; j < LL; j += 256) {
    const float val = (j < vl) ? srow_s[j] : NEGV;
    att_out[bi * LL + j] = __expf(val - M) * inv;
  }

  // unnormalized embedding accumulator:
  //   emb_pre = sum_j att_j * (q_i + rel_ij) = satt*q_i + inv * sum_j e_j*rel_ij
  if (wid == 0) {
    float ex = 0.f, ey = 0.f, ez = 0.f, ew = 0.f;
#pragma unroll
    for (int w = 0; w < 8; ++w) {
      ex += stage[w][4 * lane + 0];
      ey += stage[w][4 * lane + 1];
      ez += stage[w][4 * lane + 2];
      ew += stage[w][4 * lane + 3];
    }
    const float satt = S * inv;  // sum of att weights (== 1 up to rounding)
    float4 o;
    o.x = fmaf(satt, qi4.x, ex * inv);
    o.y = fmaf(satt, qi4.y, ey * inv);
    o.z = fmaf(satt, qi4.z, ez * inv);
    o.w = fmaf(satt, qi4.w, ew * inv);
    reinterpret_cast<float4*>(embacc + bi * DD)[lane] = o;
  }
}

// ---------------------------------------------------------------------------
// Kernel 3: deterministic reduction of per-row norm partials -> global sumsq
// ---------------------------------------------------------------------------
__global__ __launch_bounds__(256) void gsum_kernel(const float* __restrict__ n2part,
                                                   float* __restrict__ gsum) {
  __shared__ float s_s[256];
  const int tid = threadIdx.x;
  float t = 0.f;
  for (int k = tid; k < BB * LL; k += 256) t += n2part[k];
  s_s[tid] = t;
  __syncthreads();
  for (int off = 128; off > 0; off >>= 1) {
    if (tid < off) s_s[tid] += s_s[tid + off];
    __syncthreads();
  }
  if (tid == 0) gsum[0] = s_s[0];
}

// ---------------------------------------------------------------------------
// Kernel 4: epilogue over [B*L] rows: apply global scale, l2n, residual, l2n.
// ---------------------------------------------------------------------------
__global__ __launch_bounds__(256) void final_norm_kernel(
    const float* __restrict__ embacc, const float* __restrict__ v2,
    const float* __restrict__ gsum,   float* __restrict__ out_emb) {
  const int wid = threadIdx.x >> 5, lane = threadIdx.x & 31;
  const int row = blockIdx.x * 8 + wid;
  if (row >= BB * LL) return;
  const float gscale = rsqrtf(fmaxf(gsum[0], EPSV));

  float4 e = reinterpret_cast<const float4*>(embacc + (size_t)row * DD)[lane];
  e.x *= gscale; e.y *= gscale; e.z *= gscale; e.w *= gscale;
  const float n  = wave_sum(e.x * e.x + e.y * e.y + e.z * e.z + e.w * e.w);
  const float rs = rsqrtf(fmaxf(n, EPSV));
  e.x *= rs; e.y *= rs; e.z *= rs; e.w *= rs;

  const float4 v4 = reinterpret_cast<const float4*>(v2 + (size_t)row * DD)[lane];
  const float ox = v4.x + e.x, oy = v4.y + e.y, oz = v4.z + e.z, ow = v4.w + e.w;
  const float n2  = wave_sum(ox * ox + oy * oy + oz * oz + ow * ow);
  const float rs2 = rsqrtf(fmaxf(n2, EPSV));
  float4 o;
  o.x = ox * rs2; o.y = oy * rs2; o.z = oz * rs2; o.w = ow * rs2;
  reinterpret_cast<float4*>(out_emb + (size_t)row * DD)[lane] = o;
}

// ---------------------------------------------------------------------------
extern "C" void kernel_launch(void* const* d_in, const int* in_sizes, int n_in,
                              void* d_out, int out_size, void* d_ws, size_t ws_size,
                              hipStream_t stream) {
  (void)in_sizes; (void)n_in; (void)out_size; (void)ws_size;

  const float* X     = (const float*)d_in[0];
  const float* rel   = (const float*)d_in[1];
  const int*   vlens = (const int*)d_in[2];
  const float* Wq    = (const float*)d_in[3];
  const float* bq    = (const float*)d_in[4];
  // d_in[5] = Wk, d_in[6] = bk: computed in reference but unused in rel path
  const float* Wv    = (const float*)d_in[7];
  const float* bv    = (const float*)d_in[8];

  float* out     = (float*)d_out;
  float* emb_out = out;                       // [B,L,D]
  float* att_out = out + BB * LL * DD;        // [B,L,L]
  float* sc_out  = att_out + BB * LL * LL;    // [B,L,L]

  float* ws     = (float*)d_ws;
  float* gsum   = ws;                         // [16]  (1 used)
  float* qbuf   = ws + 16;                    // [B*L*D]
  float* vbuf   = qbuf + BB * LL * DD;        // [B*L*D] (contiguous with qbuf)
  float* n2part = vbuf + BB * LL * DD;        // [B*L]
  float* embacc = n2part + BB * LL;           // [B*L*D]

  // 1) projections via WMMA
  proj_wmma_kernel<<<dim3(DD / 16, (BB * LL) / 16), 32, 0, stream>>>(
      X, Wq, bq, Wv, bv, qbuf, vbuf);
  // 2) l2-normalize q and v rows (contiguous 2*B*L rows)
  l2n_rows_kernel<<<(2 * BB * LL) / 8, 256, 0, stream>>>(qbuf, 2 * BB * LL);
  // 3) single streaming pass over rel: scores + online softmax + att + emb acc
  fused_attn_kernel<<<dim3(LL, BB), 256, 0, stream>>>(
      qbuf, rel, vlens, sc_out, att_out, n2part, embacc);
  // 4) global sum-of-squares (deterministic)
  gsum_kernel<<<1, 256, 0, stream>>>(n2part, gsum);
  // 5) tiny epilogue: global scale + l2n + residual + l2n
  final_norm_kernel<<<(BB * LL) / 8, 256, 0, stream>>>(
      embacc, vbuf, gsum, emb_out);
}